// ann6_snn2_69621419868182
// MI455X (gfx1250) — compile-verified
//
#include <hip/hip_runtime.h>
#include <hip/hip_bf16.h>
#include <math.h>

typedef float v2f __attribute__((ext_vector_type(2)));
typedef float v8f __attribute__((ext_vector_type(8)));
typedef int   v4i __attribute__((ext_vector_type(4)));

#define NCH   53824      // C*H*W = 64*29*29
#define TT    50
#define BB    16
#define MM    (TT*BB)    // 800 GEMM rows
#define NOUT7 512
#define ALPHA_LIF 0.77880078307140486825f   // exp(-1/4)

#ifndef __has_builtin
#define __has_builtin(x) 0
#endif

#if defined(__HIP_DEVICE_COMPILE__) && __has_builtin(__builtin_amdgcn_global_load_async_to_lds_b128)
#define ASYNC_LDS 1
#else
#define ASYNC_LDS 0
#endif

#if ASYNC_LDS
typedef v4i __attribute__((address_space(1))) g_v4i;   // global (AS1)
typedef v4i __attribute__((address_space(3))) l_v4i;   // LDS (AS3)
__device__ __forceinline__ void async_cp16(const float* g, float* l) {
  __builtin_amdgcn_global_load_async_to_lds_b128((g_v4i*)g, (l_v4i*)l, 0, 0);
}
__device__ __forceinline__ void async_wait0() {
#if __has_builtin(__builtin_amdgcn_s_wait_asynccnt)
  __builtin_amdgcn_s_wait_asynccnt(0);
#else
  asm volatile("s_wait_asynccnt 0" ::: "memory");
#endif
}
#endif

// ---------------------------------------------------------------------------
// Kernel 1: sigmoid + transpose  [B, n, T] -> xs[m = t*B+b, i]   (coalesced)
// ---------------------------------------------------------------------------
__global__ __launch_bounds__(256) void k_sig_transpose(
    const float* __restrict__ in, float* __restrict__ xs) {
  __shared__ __align__(16) float tile[64][52];
  const int b  = blockIdx.y;
  const int i0 = blockIdx.x * 64;          // 841 blocks of 64 channels
  const int tid = threadIdx.x;
  const size_t base = ((size_t)b * NCH + i0) * TT;   // contiguous 64*50 region
  for (int l = tid; l < 64 * TT; l += 256) {
    float v = in[base + l];                          // fully coalesced read
    tile[l / TT][l % TT] = 1.0f / (1.0f + expf(-v)); // sigmoid
  }
  __syncthreads();
  for (int l = tid; l < 64 * TT; l += 256) {
    int tt = l >> 6, ii = l & 63;                    // ii fastest -> coalesced
    xs[(size_t)(tt * BB + b) * NCH + (i0 + ii)] = tile[ii][tt];
  }
}

// ---------------------------------------------------------------------------
// Kernel 2: GEMM1  g7[800,512] = xs[800,53824] @ w7[512,53824]^T
// 4 waves/block (wave32), 2x2 wave grid over a 32x64 macro-tile.
// Each wave: 16x32 tile = TWO independent f32 accumulators sharing the same
// A registers (halves LDS A-reads, breaks the WMMA RAW chain).
// LDS staging via GLOBAL_LOAD_ASYNC_TO_LDS_B128 (ASYNCcnt-tracked).
// ---------------------------------------------------------------------------
__global__ __launch_bounds__(128) void k_gemm1_wmma(
    const float* __restrict__ xs, const float* __restrict__ w7,
    float* __restrict__ g7) {
  __shared__ __align__(16) float As[32][36];   // stride 36 keeps 16B align
  __shared__ __align__(16) float Bs[64][36];
  const int tid  = threadIdx.x;
  const int m0   = blockIdx.y * 32;
  const int n0   = blockIdx.x * 64;
  const int wave = tid >> 5, lane = tid & 31;
  const int wm   = (wave & 1) * 16;            // wave tile row offset
  const int wn   = (wave >> 1) * 32;           // wave tile col offset (16x32)
  const int r    = lane & 15;
  const int klo  = (lane >> 4) * 2;            // K sub-offset per half-wave
  const int lrow = tid >> 3;                   // 0..15 cooperative-load row
  const int lc4  = (tid & 7) * 4;              // float4 column
  v8f acc0 = {}, acc1 = {};

  const float* gA0 = xs + (size_t)(m0 + lrow) * NCH + lc4;
  const float* gA1 = xs + (size_t)(m0 + 16 + lrow) * NCH + lc4;
  const float* gB0 = w7 + (size_t)(n0 + lrow) * NCH + lc4;
  const float* gB1 = w7 + (size_t)(n0 + 16 + lrow) * NCH + lc4;
  const float* gB2 = w7 + (size_t)(n0 + 32 + lrow) * NCH + lc4;
  const float* gB3 = w7 + (size_t)(n0 + 48 + lrow) * NCH + lc4;

  for (int k0 = 0; k0 < NCH; k0 += 32) {
#if ASYNC_LDS
    __syncthreads();                           // compute on old stage finished
    async_cp16(gA0 + k0, &As[lrow][lc4]);
    async_cp16(gA1 + k0, &As[16 + lrow][lc4]);
    async_cp16(gB0 + k0, &Bs[lrow][lc4]);
    async_cp16(gB1 + k0, &Bs[16 + lrow][lc4]);
    async_cp16(gB2 + k0, &Bs[32 + lrow][lc4]);
    async_cp16(gB3 + k0, &Bs[48 + lrow][lc4]);
    __builtin_prefetch(gA0 + k0 + 128, 0, 0);  // global_prefetch_b8 (L2)
    __builtin_prefetch(gB0 + k0 + 128, 0, 0);
    async_wait0();                             // own async writes landed
    __syncthreads();                           // all waves' writes visible
#else
    float4 av0 = *(const float4*)(gA0 + k0);
    float4 av1 = *(const float4*)(gA1 + k0);
    float4 bv0 = *(const float4*)(gB0 + k0);
    float4 bv1 = *(const float4*)(gB1 + k0);
    float4 bv2 = *(const float4*)(gB2 + k0);
    float4 bv3 = *(const float4*)(gB3 + k0);
    __builtin_prefetch(gA0 + k0 + 128, 0, 0);
    __builtin_prefetch(gB0 + k0 + 128, 0, 0);
    __syncthreads();
    *(float4*)(&As[lrow][lc4])      = av0;
    *(float4*)(&As[16 + lrow][lc4]) = av1;
    *(float4*)(&Bs[lrow][lc4])      = bv0;
    *(float4*)(&Bs[16 + lrow][lc4]) = bv1;
    *(float4*)(&Bs[32 + lrow][lc4]) = bv2;
    *(float4*)(&Bs[48 + lrow][lc4]) = bv3;
    __syncthreads();
#endif
#pragma unroll
    for (int ks = 0; ks < 8; ++ks) {
      const int kc = ks * 4 + klo;
      v2f a, b0, b1;
      // A 16x4 f32: lanes 0-15 -> K={0,1}, lanes 16-31 -> K={2,3} (ISA 7.12.2)
      a.x  = As[wm + r][kc];          a.y  = As[wm + r][kc + 1];
      b0.x = Bs[wn + r][kc];          b0.y = Bs[wn + r][kc + 1];
      b1.x = Bs[wn + 16 + r][kc];     b1.y = Bs[wn + 16 + r][kc + 1];
      acc0 = __builtin_amdgcn_wmma_f32_16x16x4_f32(
          false, a, false, b0, (short)0, acc0, false, false);
      acc1 = __builtin_amdgcn_wmma_f32_16x16x4_f32(
          false, a, false, b1, (short)0, acc1, false, false);
    }
  }
  // D layout: VGPR j = row (wm + j) for lanes 0-15, (wm + 8 + j) for 16-31
  const int dm = m0 + wm + (lane >> 4) * 8;
  const int dn = n0 + wn + (lane & 15);
#pragma unroll
  for (int j = 0; j < 8; ++j) {
    g7[(size_t)(dm + j) * NOUT7 + dn]      = acc0[j];
    g7[(size_t)(dm + j) * NOUT7 + dn + 16] = acc1[j];
  }
}

// ---------------------------------------------------------------------------
// Kernel 3: per-(b,o) time scan. IIR7 (commuted past the GEMM; uniform
// coefficients), + bias7, LIF7 (hard reset), then IIR8 on the spikes.
// ---------------------------------------------------------------------------
__global__ __launch_bounds__(256) void k_scan_mid(
    const float* __restrict__ g7, const float* __restrict__ b7,
    const float* __restrict__ a1_7, const float* __restrict__ a2_7,
    const float* __restrict__ a1_8, const float* __restrict__ a2_8,
    float* __restrict__ a8out) {
  const int idx = blockIdx.x * blockDim.x + threadIdx.x;
  if (idx >= BB * NOUT7) return;
  const int o = idx & (NOUT7 - 1);
  const int b = idx >> 9;
  const float c1 = a1_7[0], c2 = a2_7[0];     // uniform-filled by reference
  const float d1 = a1_8[o], d2 = a2_8[o];
  const float bias = b7[o];
  float y1 = 0.f, y2 = 0.f, v = 0.f, z1 = 0.f, z2 = 0.f;
#pragma unroll 1
  for (int t = 0; t < TT; ++t) {
    const float g = g7[(size_t)(t * BB + b) * NOUT7 + o];  // coalesced over o
    const float y = c1 * y1 + c2 * y2 + g;    // dual-exp IIR (layer 7)
    y2 = y1; y1 = y;
    v = ALPHA_LIF * v + (y + bias);           // LIF membrane
    const float s = (v > 1.0f) ? 1.0f : 0.0f; // spike
    v = (v > 1.0f) ? 0.0f : v;                // hard reset
    const float z = d1 * z1 + d2 * z2 + s;    // dual-exp IIR (layer 8)
    z2 = z1; z1 = z;
    a8out[(size_t)(t * BB + b) * NOUT7 + o] = z;
  }
}

// ---------------------------------------------------------------------------
// Kernel 4: GEMM2  i8p[800,16] = a8[800,512] @ w8[10,512]^T  (N padded to 16)
// One wave per 16-row tile; 128 x V_WMMA_F32_16X16X4_F32 per wave.
// ---------------------------------------------------------------------------
__global__ __launch_bounds__(32) void k_gemm2_wmma(
    const float* __restrict__ a8, const float* __restrict__ w8,
    float* __restrict__ i8p) {
  const int lane = threadIdx.x & 31;
  const int m0   = blockIdx.x * 16;
  const int r    = lane & 15;
  const int klo  = (lane >> 4) * 2;
  const float mask = (r < 10) ? 1.0f : 0.0f;            // branchless pad
  const float* wrow = w8 + (size_t)(r < 10 ? r : 9) * NOUT7;
  const float* arow = a8 + (size_t)(m0 + r) * NOUT7;
  v8f acc = {};
#pragma unroll 4
  for (int k0 = 0; k0 < NOUT7; k0 += 4) {
    v2f a, b;
    a.x = arow[k0 + klo];            a.y = arow[k0 + klo + 1];
    b.x = wrow[k0 + klo] * mask;     b.y = wrow[k0 + klo + 1] * mask;
    acc = __builtin_amdgcn_wmma_f32_16x16x4_f32(
        false, a, false, b, (short)0, acc, false, false);
  }
  const int dm = m0 + (lane >> 4) * 8;
#pragma unroll
  for (int j = 0; j < 8; ++j)
    i8p[(size_t)(dm + j) * 16 + (lane & 15)] = acc[j];
}

// ---------------------------------------------------------------------------
// Kernel 5: layer-8 bias + LIF scan, write out[b,10,t]
// ---------------------------------------------------------------------------
__global__ __launch_bounds__(256) void k_scan_out(
    const float* __restrict__ i8p, const float* __restrict__ b8,
    float* __restrict__ out) {
  const int idx = blockIdx.x * blockDim.x + threadIdx.x;
  if (idx >= BB * 10) return;
  const int o = idx % 10, b = idx / 10;
  const float bias = b8[o];
  float v = 0.f;
#pragma unroll 1
  for (int t = 0; t < TT; ++t) {
    v = ALPHA_LIF * v + (i8p[(size_t)(t * BB + b) * 16 + o] + bias);
    const float s = (v > 1.0f) ? 1.0f : 0.0f;
    v = (v > 1.0f) ? 0.0f : v;
    out[(size_t)(b * 10 + o) * TT + t] = s;
  }
}

// ---------------------------------------------------------------------------
extern "C" void kernel_launch(void* const* d_in, const int* in_sizes, int n_in,
                              void* d_out, int out_size, void* d_ws, size_t ws_size,
                              hipStream_t stream) {
  (void)in_sizes; (void)n_in; (void)out_size; (void)ws_size;
  const float* inputs = (const float*)d_in[0];
  const float* a1_7   = (const float*)d_in[1];
  const float* a2_7   = (const float*)d_in[2];
  const float* w7     = (const float*)d_in[3];
  const float* b7     = (const float*)d_in[4];
  const float* a1_8   = (const float*)d_in[5];
  const float* a2_8   = (const float*)d_in[6];
  const float* w8     = (const float*)d_in[7];
  const float* b8     = (const float*)d_in[8];
  float* out = (float*)d_out;

  // workspace carve-up (floats): xs 800*53824 | g7 800*512 | a8 800*512 | i8p 800*16
  float* xs  = (float*)d_ws;
  float* g7  = xs  + (size_t)MM * NCH;
  float* a8  = g7  + (size_t)MM * NOUT7;
  float* i8p = a8  + (size_t)MM * NOUT7;

  dim3 g1(NCH / 64, BB);                 // 841 x 16
  k_sig_transpose<<<g1, 256, 0, stream>>>(inputs, xs);

  dim3 g2(NOUT7 / 64, MM / 32);          // 8 x 25 = 200 blocks
  k_gemm1_wmma<<<g2, 128, 0, stream>>>(xs, w7, g7);

  k_scan_mid<<<(BB * NOUT7) / 256, 256, 0, stream>>>(g7, b7, a1_7, a2_7,
                                                     a1_8, a2_8, a8);

  k_gemm2_wmma<<<MM / 16, 32, 0, stream>>>(a8, w8, i8p);

  k_scan_out<<<1, 256, 0, stream>>>(i8p, b8, out);
}